// CustomPool_55241869361671
// MI455X (gfx1250) — compile-verified
//
#include <hip/hip_runtime.h>

// -------------------------------------------------------------------------
// NHWC 2x2 stride-2 max pool, fp32.
//   in : (32, 112, 112, 128)  -> out : (32, 56, 56, 128)
//
// Pure streaming kernel (memory-bound ~1000x: 257 MB traffic vs 38.5M
// max-ops; roofline floor ~11 us @ 23.3 TB/s).
//
//   - 3D grid factors the domain exactly -> zero div/mod in addressing
//   - threadIdx.x = c4 (32 lanes = one pixel's 128 channels = one wave32)
//     -> every load/store is a single coalesced 512 B/wave transaction
//   - 4x global_load_b128 + 1x global_store_b128 per thread, all NT:
//     data is touched exactly once and the 205 MB input exceeds the
//     192 MB L2, so non-temporal hints avoid pointless cache churn.
// -------------------------------------------------------------------------

typedef __attribute__((ext_vector_type(4))) float f32x4;

#define IN_N  32
#define IN_H  112
#define IN_W  112
#define CH    128
#define C4    (CH / 4)          // 32 float4 per pixel
#define OUT_H (IN_H / 2)        // 56
#define OUT_W (IN_W / 2)        // 56
#define OW_PER_BLOCK 8          // 8 output pixels per block (256 threads)

__device__ __forceinline__ f32x4 vmax4(f32x4 a, f32x4 b) {
    f32x4 r;
    r.x = fmaxf(a.x, b.x);
    r.y = fmaxf(a.y, b.y);
    r.z = fmaxf(a.z, b.z);
    r.w = fmaxf(a.w, b.w);
    return r;
}

__global__ __launch_bounds__(256) void
maxpool2x2_nhwc_kernel(const f32x4* __restrict__ in, f32x4* __restrict__ out) {
    // Division-free decomposition via the launch geometry:
    const int c4 = threadIdx.x;                              // 0..31
    const int ow = blockIdx.x * OW_PER_BLOCK + threadIdx.y;  // 0..55
    const int oh = blockIdx.y;                               // 0..55
    const int n  = blockIdx.z;                               // 0..31

    const int row_stride = IN_W * C4;   // 3584 float4 per input row

    const int in_base  = ((n * IN_H + 2 * oh) * IN_W + 2 * ow) * C4 + c4;
    const int out_idx  = ((n * OUT_H + oh) * OUT_W + ow) * C4 + c4;

    // Four window taps: (2oh,2ow), (2oh,2ow+1), (2oh+1,2ow), (2oh+1,2ow+1).
    const f32x4 a = __builtin_nontemporal_load(in + in_base);
    const f32x4 b = __builtin_nontemporal_load(in + in_base + C4);
    const f32x4 c = __builtin_nontemporal_load(in + in_base + row_stride);
    const f32x4 d = __builtin_nontemporal_load(in + in_base + row_stride + C4);

    const f32x4 m = vmax4(vmax4(a, b), vmax4(c, d));

    __builtin_nontemporal_store(m, out + out_idx);
}

extern "C" void kernel_launch(void* const* d_in, const int* in_sizes, int n_in,
                              void* d_out, int out_size, void* d_ws, size_t ws_size,
                              hipStream_t stream) {
    (void)in_sizes; (void)n_in; (void)out_size; (void)d_ws; (void)ws_size;

    const f32x4* in  = (const f32x4*)d_in[0];
    f32x4*       out = (f32x4*)d_out;

    // Exact cover of the output domain: 7*56*32 = 12,544 blocks of 256.
    dim3 block(C4, OW_PER_BLOCK, 1);                 // 32 x 8 = 256 (8 waves)
    dim3 grid(OUT_W / OW_PER_BLOCK, OUT_H, IN_N);    // (7, 56, 32)

    maxpool2x2_nhwc_kernel<<<grid, block, 0, stream>>>(in, out);
}